// MixedMSEPowerImbalance_10900626998069
// MI455X (gfx1250) — compile-verified
//
#include <hip/hip_runtime.h>
#include <math.h>

// ---------------------------------------------------------------------------
// MixedMSEPowerImbalance for MI455X (gfx1250, wave32)
// Memory-bound pipeline: node build -> edge scatter -> node finalize ->
// single-wave WMMA (f32 16x16x4) deterministic tail reduction.
// ---------------------------------------------------------------------------

typedef __attribute__((ext_vector_type(2))) float v2f;
typedef __attribute__((ext_vector_type(8))) float v8f;

#define ALPHA_C      0.9f
#define PHYS_SCALE_C 0.02f
#define INV_SN_MVA   0.01f
#define DEG2RAD_C    0.017453292519943295f
#define PAD          4096   // padded partial-array length (>= gridDim, mult of 64)

// --- block tree reduction into per-block partial --------------------------
__device__ __forceinline__ void blockReduceStore(float v, float* sm, float* dst) {
    int t = threadIdx.x;
    sm[t] = v;
    __syncthreads();
#pragma unroll
    for (int s = 128; s > 0; s >>= 1) {
        if (t < s) sm[t] += sm[t + s];
        __syncthreads();
    }
    if (t == 0) dst[blockIdx.x] = sm[0];
}

// --- pass 1: denorm, MSE partials, per-node (e,f) and (p,q,vm^2) ----------
__global__ void node_build_kernel(const float* __restrict__ y_pred,
                                  const float* __restrict__ y_true,
                                  const float* __restrict__ x_input,
                                  const float* __restrict__ x_mean,
                                  const float* __restrict__ x_std,
                                  const float* __restrict__ y_mean,
                                  const float* __restrict__ y_std,
                                  float2* __restrict__ ef,
                                  float4* __restrict__ pqv,
                                  float* __restrict__ partial_mse,
                                  int N, int nbus) {
    __shared__ float sm[256];
    int t = blockIdx.x * blockDim.x + threadIdx.x;
    float msep = 0.0f;
    if (t < N) {
        int i = t % nbus;  // bus index within batch element
        float2 yp2 = ((const float2*)y_pred)[t];
        float2 yt2 = ((const float2*)y_true)[t];
        float2 ys  = ((const float2*)y_std)[i];
        float2 ym  = ((const float2*)y_mean)[i];
        float yp0 = fmaf(yp2.x, ys.x, ym.x);
        float yp1 = fmaf(yp2.y, ys.y, ym.y);
        float yt0 = fmaf(yt2.x, ys.x, ym.x);
        float yt1 = fmaf(yt2.y, ys.y, ym.y);
        float d0 = yp0 - yt0, d1 = yp1 - yt1;
        msep = d0 * d0 + d1 * d1;

        float p = fmaf(x_input[t * 7 + 0], x_std[i * 7 + 0], x_mean[i * 7 + 0]) * INV_SN_MVA;
        float q = fmaf(x_input[t * 7 + 1], x_std[i * 7 + 1], x_mean[i * 7 + 1]) * INV_SN_MVA;

        float vm = yp0;
        float va = yp1 * DEG2RAD_C;
        float s, c;
        __sincosf(va, &s, &c);
        ef[t]  = make_float2(vm * c, vm * s);      // rectangular voltage per node
        pqv[t] = make_float4(p, q, vm * vm, 0.0f); // injections + |V|^2
    }
    blockReduceStore(msep, sm, partial_mse);
}

// --- pass 2: per-edge power flow, scatter-add onto src (L2-resident) ------
__global__ void edge_kernel(const int* __restrict__ edge_index,
                            const float4* __restrict__ edge_attr,
                            const float* __restrict__ edge_mean,
                            const float* __restrict__ edge_std,
                            const float2* __restrict__ ef,
                            float* __restrict__ agg,
                            int E) {
    int e = blockIdx.x * blockDim.x + threadIdx.x;
    if (e >= E) return;
    int src = edge_index[e];
    int dst = edge_index[E + e];

    float4 ea = edge_attr[e];                         // streamed b128
    float4 es = *(const float4*)edge_std;             // uniform -> scalar loads
    float4 em = *(const float4*)edge_mean;
    float Gs = fmaf(ea.x, es.x, em.x);
    float Bs = fmaf(ea.y, es.y, em.y);
    float Gm = fmaf(ea.z, es.z, em.z);
    float Bm = fmaf(ea.w, es.w, em.w);

    float2 efi = ef[src];                             // L2-hot gathers (8 MB table)
    float2 efj = ef[dst];
    float ei = efi.x, fi = efi.y, ej = efj.x, fj = efj.y;

    float Ire = Gs * ei - Bs * fi + Gm * ej - Bm * fj;
    float Iim = Gs * fi + Bs * ei + Gm * fj + Bm * ej;
    float Pji = -(ei * Ire + fi * Iim);
    float Qji = -(fi * Ire - ei * Iim);

    atomicAdd(&agg[2 * src + 0], Pji);                // global_atomic_add_f32 in L2
    atomicAdd(&agg[2 * src + 1], Qji);
}

// --- pass 3: per-node imbalance, masked partials --------------------------
__global__ void node_finalize_kernel(const float* __restrict__ agg,
                                     const float4* __restrict__ pqv,
                                     const float* __restrict__ shunt,
                                     const unsigned char* __restrict__ mask,
                                     float* __restrict__ partial_sqw,
                                     float* __restrict__ partial_w,
                                     int N, int nbus) {
    __shared__ float sm[256];
    int t = blockIdx.x * blockDim.x + threadIdx.x;
    float sqw = 0.0f, wv = 0.0f;
    if (t < N) {
        int i = t % nbus;
        float4 pq = pqv[t];
        float a0 = agg[2 * t + 0];
        float a1 = agg[2 * t + 1];
        float2 sh = ((const float2*)shunt)[i];
        float dP = -a0 + pq.x + pq.z * sh.x;
        float dQ = -a1 + pq.y - pq.z * sh.y;
        float w = mask[i] ? 1.0f : 0.0f;
        sqw = (dP * dP + dQ * dQ) * w;
        wv = w;
    }
    blockReduceStore(sqw, sm, partial_sqw);
    __syncthreads();
    blockReduceStore(wv, sm, partial_w);
}

// --- pass 4: single wave32; WMMA f32 16x16x4 tail reduction ---------------
// A = ones(16x4), so D = A*B + C makes every row of C hold running column
// sums of all B elements: one v_wmma folds 64 f32 partials, full f32 exact.
__global__ void final_reduce_kernel(const float* __restrict__ partial_mse,
                                    const float* __restrict__ partial_sqw,
                                    const float* __restrict__ partial_w,
                                    float* __restrict__ out,
                                    int count, float inv_mse_count) {
    __shared__ float red[32];
    __shared__ float sums[3];
    int lane = threadIdx.x;  // blockDim.x == 32 -> EXEC all ones for WMMA

    v2f a;
    a.x = 1.0f;
    a.y = 1.0f;

    const float* arrs[3] = {partial_mse, partial_sqw, partial_w};
    for (int k = 0; k < 3; ++k) {
        const float* vals = arrs[k];
        v8f c = {};
        for (int base = 0; base < count; base += 64) {
            v2f b;
            b.x = vals[base + 2 * lane + 0];
            b.y = vals[base + 2 * lane + 1];
            c = __builtin_amdgcn_wmma_f32_16x16x4_f32(
                    false, a, false, b, (short)0, c, false, false);
        }
        // All 16 rows of C are identical column-sum vectors -> sum of all
        // 256 elements equals 16 * total.
        float s = c[0] + c[1] + c[2] + c[3] + c[4] + c[5] + c[6] + c[7];
        red[lane] = s;
        __syncthreads();
        if (lane == 0) {
            float tot = 0.0f;
            for (int j = 0; j < 32; ++j) tot += red[j];
            sums[k] = tot * (1.0f / 16.0f);
        }
        __syncthreads();
    }

    if (lane == 0) {
        float mse  = sums[0] * inv_mse_count;
        float phys = sums[1] / sums[2];
        out[0] = ALPHA_C * mse + (1.0f - ALPHA_C) * PHYS_SCALE_C * phys;
    }
}

extern "C" void kernel_launch(void* const* d_in, const int* in_sizes, int n_in,
                              void* d_out, int out_size, void* d_ws, size_t ws_size,
                              hipStream_t stream) {
    const float*         y_pred     = (const float*)d_in[0];
    const float*         y_true     = (const float*)d_in[1];
    const float*         x_input    = (const float*)d_in[2];
    const int*           edge_index = (const int*)d_in[3];
    const float*         edge_attr  = (const float*)d_in[4];
    const float*         x_mean     = (const float*)d_in[5];
    const float*         x_std      = (const float*)d_in[6];
    const float*         y_mean     = (const float*)d_in[7];
    const float*         y_std      = (const float*)d_in[8];
    const float*         edge_mean  = (const float*)d_in[9];
    const float*         edge_std   = (const float*)d_in[10];
    const unsigned char* mask       = (const unsigned char*)d_in[11];  // jax bool = 1 byte
    const float*         shunt      = (const float*)d_in[12];

    const int N    = in_sizes[2] / 7;   // BATCH * N_BUS = 1,024,000
    const int nbus = in_sizes[11];      // 2000
    const int E    = in_sizes[4] / 4;   // 3,072,000

    // workspace layout (floats):
    //   [agg 2N][partial_mse PAD][partial_sqw PAD][partial_w PAD][ef 2N][pqv 4N]
    float*  ws          = (float*)d_ws;
    float*  agg         = ws;
    float*  partial_mse = ws + (size_t)2 * N;
    float*  partial_sqw = partial_mse + PAD;
    float*  partial_w   = partial_sqw + PAD;
    float2* ef          = (float2*)(partial_w + PAD);
    float4* pqv         = (float4*)((float*)ef + (size_t)2 * N);

    // zero agg + all three partial arrays in one contiguous memset (capture-safe)
    hipMemsetAsync(agg, 0, ((size_t)2 * N + 3 * PAD) * sizeof(float), stream);

    const int nb = (N + 255) / 256;  // 4000 blocks (<= PAD)
    const int eb = (E + 255) / 256;

    node_build_kernel<<<nb, 256, 0, stream>>>(y_pred, y_true, x_input, x_mean, x_std,
                                              y_mean, y_std, ef, pqv, partial_mse,
                                              N, nbus);
    edge_kernel<<<eb, 256, 0, stream>>>(edge_index, (const float4*)edge_attr,
                                        edge_mean, edge_std, ef, agg, E);
    node_finalize_kernel<<<nb, 256, 0, stream>>>(agg, pqv, shunt, mask,
                                                 partial_sqw, partial_w, N, nbus);
    final_reduce_kernel<<<1, 32, 0, stream>>>(partial_mse, partial_sqw, partial_w,
                                              (float*)d_out, PAD,
                                              1.0f / (float)((size_t)N * 2));
}